// cSigmoid_34007551049881
// MI455X (gfx1250) — compile-verified
//
#include <hip/hip_runtime.h>

// ---------------------------------------------------------------------------
// Elementwise piecewise-linear LUT (65-knot sigmoid table) over 67.1M f32.
// Memory-bound: 512 MB @ 23.3 TB/s ~= 23 us floor.
//   * b128 non-temporal streaming loads/stores (dataset is 2.7x the 192MB L2)
//   * async-to-LDS (gfx1250 data mover) table stage + s_wait_asynccnt
//   * 66-entry extended LDS coeff table folds ALL range handling:
//       entry 0 = (0,0) -> 0 below table; entry 65 = (0,1) -> 1 above table;
//       entry 64 carries the reference's "+1 for x>=xs[63]" bias.
//   * 5 VALU + 1 ds_load_b64 per element (budget ~14 VALU/elem at 23.3 TB/s)
// ---------------------------------------------------------------------------

typedef float v4f __attribute__((ext_vector_type(4)));
typedef float v2f __attribute__((ext_vector_type(2)));

#define TPB    256
#define UNROLL 4
#define NPTS   65

#if defined(__gfx1250__) && __has_builtin(__builtin_amdgcn_global_load_async_to_lds_b32)
#define HAVE_ASYNC_LDS 1
#else
#define HAVE_ASYNC_LDS 0
#endif

#if HAVE_ASYNC_LDS
typedef __attribute__((address_space(1))) int gas_int;   // global addrspace
typedef __attribute__((address_space(3))) int las_int;   // LDS addrspace
#endif

// e = clamp(floor(fma(x, inv_h, c1)), 0, 65), c1 = 1 - x0*inv_h.
// Entry e of the extended table gives (a,b); result = fma(a, x, b).
__device__ __forceinline__ float lut_eval(float xv, float inv_h, float c1,
                                          const v2f* ab2_s)
{
    float r  = fmaf(xv, inv_h, c1);
    float rc = __builtin_amdgcn_fmed3f(r, 0.0f, (float)(NPTS + 0));  // [0,65]
    int   e  = (int)rc;                       // trunc == floor (non-negative)
    v2f   ab = ab2_s[e];
    return fmaf(ab.x, xv, ab.y);
}

__global__ __launch_bounds__(TPB) void cSigmoid_lut_kernel(
    const float* __restrict__ x, const float* __restrict__ table,
    float* __restrict__ out, long long n)
{
    __shared__ float tab_s[2 * NPTS];   // raw table (xs,ys interleaved)
    __shared__ v2f   ab2_s[NPTS + 1];   // extended (slope,intercept) table

    const int t = threadIdx.x;

    // ---- stage table into LDS via gfx1250 async data mover ----
    if (t < 2 * NPTS) {
#if HAVE_ASYNC_LDS
        __builtin_amdgcn_global_load_async_to_lds_b32(
            (gas_int*)(table + t), (las_int*)(&tab_s[t]), 0, 0);
#else
        tab_s[t] = table[t];
#endif
    }
#if HAVE_ASYNC_LDS
#if __has_builtin(__builtin_amdgcn_s_wait_asynccnt)
    __builtin_amdgcn_s_wait_asynccnt(0);
#else
    asm volatile("s_wait_asynccnt 0" ::: "memory");
#endif
#endif
    __syncthreads();

    // ---- build extended coefficient table ----
    if (t < NPTS - 1) {
        float xs0 = tab_s[2 * t],     ys0 = tab_s[2 * t + 1];
        float xs1 = tab_s[2 * t + 2], ys1 = tab_s[2 * t + 3];
        float a = (ys1 - ys0) / (xs1 - xs0);
        float b = ys0 - a * xs0;
        if (t == NPTS - 2) b += 1.0f;   // fold reference's "+1 for x>=xs[n-2]"
        v2f ab; ab.x = a; ab.y = b;
        ab2_s[t + 1] = ab;
    } else if (t == NPTS - 1) {         // below-table sentinel -> 0
        v2f ab; ab.x = 0.0f; ab.y = 0.0f;
        ab2_s[0] = ab;
    } else if (t == NPTS) {             // above-table sentinel -> 1
        v2f ab; ab.x = 0.0f; ab.y = 1.0f;
        ab2_s[NPTS] = ab;
    }
    __syncthreads();

    const float x0    = tab_s[0];
    const float inv_h = (float)(NPTS - 1) / (tab_s[2 * (NPTS - 1)] - tab_s[0]);
    const float c1    = 1.0f - x0 * inv_h;

    // ---- main streaming loop ----
    const long long n4   = n >> 2;
    const long long tile = (long long)blockIdx.x * (TPB * UNROLL);
    const v4f* __restrict__ xin = (const v4f*)x;
    v4f*       __restrict__ xo  = (v4f*)out;

    if (tile + TPB * UNROLL <= n4) {
        // fast path: 4 unconditional b128 NT loads -> compute -> 4 NT stores
        v4f v[UNROLL];
#pragma unroll
        for (int k = 0; k < UNROLL; ++k)
            v[k] = __builtin_nontemporal_load(xin + tile + t + k * TPB);
        v4f f[UNROLL];
#pragma unroll
        for (int k = 0; k < UNROLL; ++k)
#pragma unroll
            for (int c = 0; c < 4; ++c)
                f[k][c] = lut_eval(v[k][c], inv_h, c1, ab2_s);
#pragma unroll
        for (int k = 0; k < UNROLL; ++k)
            __builtin_nontemporal_store(f[k], xo + tile + t + k * TPB);
    } else {
        // tail block: guarded per chunk
#pragma unroll
        for (int k = 0; k < UNROLL; ++k) {
            long long i = tile + t + (long long)k * TPB;
            if (i < n4) {
                v4f v = __builtin_nontemporal_load(xin + i);
                v4f f;
#pragma unroll
                for (int c = 0; c < 4; ++c)
                    f[c] = lut_eval(v[c], inv_h, c1, ab2_s);
                __builtin_nontemporal_store(f, xo + i);
            }
        }
        // scalar tail (n not a multiple of 4)
        long long tail = n & 3LL;
        if (tile == 0 && (long long)t < tail) {
            long long i = (n4 << 2) + t;
            out[i] = lut_eval(x[i], inv_h, c1, ab2_s);
        }
    }
}

extern "C" void kernel_launch(void* const* d_in, const int* in_sizes, int n_in,
                              void* d_out, int out_size, void* d_ws, size_t ws_size,
                              hipStream_t stream)
{
    const float* x     = (const float*)d_in[0];
    const float* table = (const float*)d_in[1];
    float*       out   = (float*)d_out;

    long long n  = (long long)in_sizes[0];       // 8*4096*2048 = 67,108,864
    long long n4 = n >> 2;
    long long chunk = (long long)TPB * UNROLL;   // float4s per block
    long long grid  = (n4 + chunk - 1) / chunk;
    if (grid < 1) grid = 1;

    cSigmoid_lut_kernel<<<(dim3)(unsigned)grid, TPB, 0, stream>>>(x, table, out, n);
}